// TensorProduct_59236188946490
// MI455X (gfx1250) — compile-verified
//
#include <hip/hip_runtime.h>
#include <cstdint>

// e3nn 'uvu' tensor product: 128x0e+128x1o  (x)  1x0e+1x1o  ->  128x0e+128x1o
// Memory-bound (~0.78 flop/byte): optimize the data path, not the math.
// CDNA5 path: async Global<->LDS staging so ALL global traffic is coalesced b128.

#ifndef __has_builtin
#define __has_builtin(x) 0
#endif

#if __has_builtin(__builtin_amdgcn_global_load_async_to_lds_b128) && \
    __has_builtin(__builtin_amdgcn_global_store_async_from_lds_b128)
#define USE_ASYNC_LDS 1
#else
#define USE_ASYNC_LDS 0
#endif

typedef int v4i __attribute__((vector_size(16)));              // builtin's elem type
typedef __attribute__((address_space(3))) v4i lds_v4i;         // LDS-qualified elem

__device__ __forceinline__ lds_v4i* to_lds(const void* p) {
  // generic 'shared' pointer: low 32 bits are the LDS byte address
  return (lds_v4i*)(uint32_t)(uintptr_t)p;
}
__device__ __forceinline__ v4i* to_glb(const void* p) {
  return (v4i*)(uintptr_t)p;
}

__global__ __launch_bounds__(128)
void tp_uvu_kernel(const float* __restrict__ x1,
                   const float* __restrict__ x2,
                   const float* __restrict__ w,
                   float* __restrict__ out)
{
  const int z = blockIdx.x;      // edge
  const int u = threadIdx.x;     // channel 0..127

  const float c3 = 0.57735026918962576f;  // 1/sqrt(3)
  const float c6 = 0.40824829046386302f;  // 1/sqrt(6)

  const float* x1r  = x1  + (size_t)z * 512;
  const float* wr   = w   + (size_t)z * 640;
  float*       outr = out + (size_t)z * 512;

  // x2 row broadcast: one b128 per wave (same address -> single request)
  const float4 q = ((const float4*)x2)[z];
  const float s = q.x, vx = q.y, vy = q.z, vz = q.w;

#if USE_ASYNC_LDS
  __shared__ float sX[512];
  __shared__ float sW[640];
  __shared__ float sO[512];

  // Stage x1 row (512 f32) and weight row (640 f32) as pure b128 transactions.
  __builtin_amdgcn_global_load_async_to_lds_b128(to_glb(x1r + 4 * u),
                                                 to_lds(&sX[4 * u]), 0, 0);
  __builtin_amdgcn_global_load_async_to_lds_b128(to_glb(wr + 4 * u),
                                                 to_lds(&sW[4 * u]), 0, 0);
  if (u < 32) {
    __builtin_amdgcn_global_load_async_to_lds_b128(to_glb(wr + 512 + 4 * u),
                                                   to_lds(&sW[512 + 4 * u]), 0, 0);
  }

#if __has_builtin(__builtin_amdgcn_s_wait_asynccnt)
  __builtin_amdgcn_s_wait_asynccnt(0);
#else
  asm volatile("s_wait_asynccnt 0" ::: "memory");
#endif
  __syncthreads();

  const float a  = sX[u];
  const float bx = sX[128 + 3 * u + 0];   // bank = 3u mod 64: conflict-free
  const float by = sX[128 + 3 * u + 1];
  const float bz = sX[128 + 3 * u + 2];
  const float w0 = sW[u];
  const float w1 = sW[128 + u];
  const float w2 = sW[256 + u];
  const float w3 = sW[384 + u];
  const float w4 = sW[512 + u];
#else
  const float a  = x1r[u];
  const float bx = x1r[128 + 3 * u + 0];
  const float by = x1r[128 + 3 * u + 1];
  const float bz = x1r[128 + 3 * u + 2];
  const float w0 = wr[u];
  const float w1 = wr[128 + u];
  const float w2 = wr[256 + u];
  const float w3 = wr[384 + u];
  const float w4 = wr[512 + u];
#endif

  // out0[u] = w0*a*s + (1/sqrt3)*w3*(b.v)
  const float dotbv = bx * vx + by * vy + bz * vz;
  const float o0 = w0 * a * s + c3 * w3 * dotbv;

  // out1[u,k] = (1/sqrt3)*w1*a*v[k] + (1/sqrt3)*w2*s*b[k] + (1/sqrt6)*w4*(b x v)[k]
  const float t1 = c3 * w1 * a;
  const float t2 = c3 * w2 * s;
  const float t4 = c6 * w4;
  const float ox = t1 * vx + t2 * bx + t4 * (by * vz - bz * vy);
  const float oy = t1 * vy + t2 * by + t4 * (bz * vx - bx * vz);
  const float oz = t1 * vz + t2 * bz + t4 * (bx * vy - by * vx);

#if USE_ASYNC_LDS
  sO[u] = o0;
  sO[128 + 3 * u + 0] = ox;
  sO[128 + 3 * u + 1] = oy;
  sO[128 + 3 * u + 2] = oz;
  __syncthreads();

  // Stream the output row back as pure b128 transactions.
  __builtin_amdgcn_global_store_async_from_lds_b128(to_glb(outr + 4 * u),
                                                    to_lds(&sO[4 * u]), 0, 0);
  // S_ENDPGM performs an implicit wait-idle, covering outstanding ASYNCcnt.
#else
  outr[u] = o0;
  outr[128 + 3 * u + 0] = ox;
  outr[128 + 3 * u + 1] = oy;
  outr[128 + 3 * u + 2] = oz;
#endif
}

extern "C" void kernel_launch(void* const* d_in, const int* in_sizes, int n_in,
                              void* d_out, int out_size, void* d_ws, size_t ws_size,
                              hipStream_t stream) {
  const float* x1 = (const float*)d_in[0];   // (n, 512) f32
  const float* x2 = (const float*)d_in[1];   // (n, 4)   f32
  const float* w  = (const float*)d_in[2];   // (n, 640) f32
  float* out = (float*)d_out;                // (n, 512) f32

  const int n = in_sizes[0] / 512;           // 131072 edges
  dim3 grid((unsigned)n), block(128);
  hipLaunchKernelGGL(tp_uvu_kernel, grid, block, 0, stream, x1, x2, w, out);
}